// MultiSimilarityLoss_56839597195693
// MI455X (gfx1250) — compile-verified
//
#include <hip/hip_runtime.h>
#include <math.h>

typedef __attribute__((ext_vector_type(2))) float v2f;
typedef __attribute__((ext_vector_type(8))) float v8f;

namespace {
constexpr float kAlpha = 2.0f;
constexpr float kBeta  = 50.0f;
constexpr float kLamda = 1.0f;
constexpr float kEps   = 0.1f;
constexpr int   kB     = 8192;   // batch
constexpr int   kD     = 128;    // embedding dim
constexpr int   kWaves = 8;      // 256 threads / wave32
constexpr int   kTiles = kB / 16;  // 512 column tiles (and row blocks)
}

// ---------------------------------------------------------------------------
// Pass 1: per-row mining thresholds min_pos / max_neg via fused WMMA GEMM.
// Block = 16 rows; 8 waves sweep interleaved 16-column tiles.
// ---------------------------------------------------------------------------
__global__ __launch_bounds__(256) void ms_thresholds(const float* __restrict__ x,
                                                     const int* __restrict__ y,
                                                     float* __restrict__ minpos,
                                                     float* __restrict__ maxneg) {
  const int i0   = blockIdx.x * 16;
  const int wave = threadIdx.x >> 5;
  const int lane = threadIdx.x & 31;
  const int ln   = lane & 15;          // N (or M for A frag) index
  const int kk   = (lane >> 4) << 1;   // K sub-offset {0,2} per half-wave
  const int hi   = (lane >> 4) << 3;   // row offset {0,8} for C layout

  // Hoisted A fragments: rows i0..i0+15, reused for all 64 column tiles.
  v2f afrag[32];
  const float* arow = x + (size_t)(i0 + ln) * kD + kk;
#pragma unroll
  for (int s = 0; s < 32; ++s) afrag[s] = *(const v2f*)(arow + 4 * s);

  int yr[8];
#pragma unroll
  for (int v = 0; v < 8; ++v) yr[v] = y[i0 + hi + v];

  float rmin[8], rmax[8];
#pragma unroll
  for (int v = 0; v < 8; ++v) { rmin[v] = __builtin_inff(); rmax[v] = -__builtin_inff(); }

  for (int t = wave; t < kTiles; t += kWaves) {
    const int j0 = t * 16;
    const float* brow = x + (size_t)(j0 + ln) * kD + kk;
    const int yc = y[j0 + ln];
    v8f c = {0.f, 0.f, 0.f, 0.f, 0.f, 0.f, 0.f, 0.f};
#pragma unroll
    for (int s = 0; s < 32; ++s) {
      v2f b = *(const v2f*)(brow + 4 * s);
      c = __builtin_amdgcn_wmma_f32_16x16x4_f32(false, afrag[s], false, b,
                                                (short)0, c, false, false);
    }
    const int j = j0 + ln;
#pragma unroll
    for (int v = 0; v < 8; ++v) {
      const float sv  = c[v];
      const int   i   = i0 + hi + v;
      const bool same = (yr[v] == yc);
      if (same && (i != j)) rmin[v] = fminf(rmin[v], sv);   // positives
      if (!same)            rmax[v] = fmaxf(rmax[v], sv);   // negatives
    }
  }

  // Reduce over N: 16 lanes of each half-wave hold distinct columns.
#pragma unroll
  for (int m = 1; m < 16; m <<= 1) {
#pragma unroll
    for (int v = 0; v < 8; ++v) {
      rmin[v] = fminf(rmin[v], __shfl_xor(rmin[v], m, 32));
      rmax[v] = fmaxf(rmax[v], __shfl_xor(rmax[v], m, 32));
    }
  }
  __shared__ float smin[kWaves][16];
  __shared__ float smax[kWaves][16];
  if (ln == 0) {
#pragma unroll
    for (int v = 0; v < 8; ++v) { smin[wave][hi + v] = rmin[v]; smax[wave][hi + v] = rmax[v]; }
  }
  __syncthreads();
  if (threadIdx.x < 16) {
    float mn = __builtin_inff(), mx = -__builtin_inff();
    for (int w = 0; w < kWaves; ++w) {
      mn = fminf(mn, smin[w][threadIdx.x]);
      mx = fmaxf(mx, smax[w][threadIdx.x]);
    }
    minpos[i0 + threadIdx.x] = mn;
    maxneg[i0 + threadIdx.x] = mx;
  }
}

// ---------------------------------------------------------------------------
// Pass 2: recompute S tiles, apply keep masks, per-row exp-sums -> per-block
// partial (loss_sum, valid_count).
// ---------------------------------------------------------------------------
__global__ __launch_bounds__(256) void ms_sums(const float* __restrict__ x,
                                               const int* __restrict__ y,
                                               const float* __restrict__ minpos,
                                               const float* __restrict__ maxneg,
                                               float* __restrict__ bsum,
                                               float* __restrict__ bcnt) {
  const int i0   = blockIdx.x * 16;
  const int wave = threadIdx.x >> 5;
  const int lane = threadIdx.x & 31;
  const int ln   = lane & 15;
  const int kk   = (lane >> 4) << 1;
  const int hi   = (lane >> 4) << 3;

  v2f afrag[32];
  const float* arow = x + (size_t)(i0 + ln) * kD + kk;
#pragma unroll
  for (int s = 0; s < 32; ++s) afrag[s] = *(const v2f*)(arow + 4 * s);

  int yr[8];
  float mp[8], mx[8];
#pragma unroll
  for (int v = 0; v < 8; ++v) {
    yr[v] = y[i0 + hi + v];
    mp[v] = minpos[i0 + hi + v];
    mx[v] = maxneg[i0 + hi + v];
  }
  float psum[8], nsum[8], pcnt[8], ncnt[8];
#pragma unroll
  for (int v = 0; v < 8; ++v) { psum[v] = nsum[v] = pcnt[v] = ncnt[v] = 0.f; }

  for (int t = wave; t < kTiles; t += kWaves) {
    const int j0 = t * 16;
    const float* brow = x + (size_t)(j0 + ln) * kD + kk;
    const int yc = y[j0 + ln];
    v8f c = {0.f, 0.f, 0.f, 0.f, 0.f, 0.f, 0.f, 0.f};
#pragma unroll
    for (int s = 0; s < 32; ++s) {
      v2f b = *(const v2f*)(brow + 4 * s);
      c = __builtin_amdgcn_wmma_f32_16x16x4_f32(false, afrag[s], false, b,
                                                (short)0, c, false, false);
    }
    const int j = j0 + ln;
#pragma unroll
    for (int v = 0; v < 8; ++v) {
      const float sv  = c[v];
      const int   i   = i0 + hi + v;
      const bool same = (yr[v] == yc);
      const bool pos  = same && (i != j);
      const bool neg  = !same;
      if (neg && (sv + kEps > mp[v])) { nsum[v] += __expf(kBeta * (sv - kLamda));  ncnt[v] += 1.f; }
      if (pos && (sv - kEps < mx[v])) { psum[v] += __expf(-kAlpha * (sv - kLamda)); pcnt[v] += 1.f; }
    }
  }

#pragma unroll
  for (int m = 1; m < 16; m <<= 1) {
#pragma unroll
    for (int v = 0; v < 8; ++v) {
      psum[v] += __shfl_xor(psum[v], m, 32);
      nsum[v] += __shfl_xor(nsum[v], m, 32);
      pcnt[v] += __shfl_xor(pcnt[v], m, 32);
      ncnt[v] += __shfl_xor(ncnt[v], m, 32);
    }
  }
  __shared__ float sps[kWaves][16], sns[kWaves][16], spc[kWaves][16], snc[kWaves][16];
  if (ln == 0) {
#pragma unroll
    for (int v = 0; v < 8; ++v) {
      sps[wave][hi + v] = psum[v]; sns[wave][hi + v] = nsum[v];
      spc[wave][hi + v] = pcnt[v]; snc[wave][hi + v] = ncnt[v];
    }
  }
  __syncthreads();
  __shared__ float sper[16], sval[16];
  if (threadIdx.x < 16) {
    float ps = 0.f, ns = 0.f, pc = 0.f, nc = 0.f;
    for (int w = 0; w < kWaves; ++w) {
      ps += sps[w][threadIdx.x]; ns += sns[w][threadIdx.x];
      pc += spc[w][threadIdx.x]; nc += snc[w][threadIdx.x];
    }
    const bool valid = (pc > 0.f) && (nc > 0.f);
    sper[threadIdx.x] = valid ? ((1.f / kAlpha) * log1pf(ps) + (1.f / kBeta) * log1pf(ns)) : 0.f;
    sval[threadIdx.x] = valid ? 1.f : 0.f;
  }
  __syncthreads();
  if (threadIdx.x == 0) {
    float s = 0.f, cvd = 0.f;
    for (int r = 0; r < 16; ++r) { s += sper[r]; cvd += sval[r]; }
    bsum[blockIdx.x] = s;
    bcnt[blockIdx.x] = cvd;
  }
}

// ---------------------------------------------------------------------------
// Pass 3: deterministic 512-way tree reduction -> scalar mean loss.
// ---------------------------------------------------------------------------
__global__ __launch_bounds__(512) void ms_finalize(const float* __restrict__ bsum,
                                                   const float* __restrict__ bcnt,
                                                   float* __restrict__ out) {
  __shared__ float ss[512], sc[512];
  const int t = threadIdx.x;
  ss[t] = bsum[t];
  sc[t] = bcnt[t];
  __syncthreads();
  for (int off = 256; off > 0; off >>= 1) {
    if (t < off) { ss[t] += ss[t + off]; sc[t] += sc[t + off]; }
    __syncthreads();
  }
  if (t == 0) out[0] = (sc[0] > 0.f) ? ss[0] / sc[0] : 0.f;
}

extern "C" void kernel_launch(void* const* d_in, const int* in_sizes, int n_in,
                              void* d_out, int out_size, void* d_ws, size_t ws_size,
                              hipStream_t stream) {
  (void)in_sizes; (void)n_in; (void)out_size; (void)ws_size;
  const float* x = (const float*)d_in[0];
  const int*   y = (const int*)d_in[1];

  float* ws     = (float*)d_ws;
  float* minpos = ws;                 // [kB]
  float* maxneg = ws + kB;            // [kB]
  float* bsum   = ws + 2 * kB;        // [kTiles]
  float* bcnt   = bsum + kTiles;      // [kTiles]

  ms_thresholds<<<kTiles, 256, 0, stream>>>(x, y, minpos, maxneg);
  ms_sums<<<kTiles, 256, 0, stream>>>(x, y, minpos, maxneg, bsum, bcnt);
  ms_finalize<<<1, 512, 0, stream>>>(bsum, bcnt, (float*)d_out);
}